// BRIGHT_A_33878702031061
// MI455X (gfx1250) — compile-verified
//
#include <hip/hip_runtime.h>
#include <hip/hip_bf16.h>

typedef float v2f __attribute__((ext_vector_type(2)));
typedef float v8f __attribute__((ext_vector_type(8)));

#define LDSA 132  // 128 + 4 pad -> conflict-free LDS reads of A fragments

// ---------------------------------------------------------------------------
// Y[N,128] = X[N,128] @ W[wrowoff:wrowoff+128, 0:128] (+ bias) (+= Y if accum)
// One block = 16 rows x 128 cols, 8 waves, wave w owns cols [16w,16w+16).
// Uses V_WMMA_F32_16X16X4_F32, K-loop of 32 steps. B addresses are constant
// immediate offsets from a single per-thread base pointer (24-bit IOFFSET).
// ---------------------------------------------------------------------------
__global__ __launch_bounds__(256)
void ga_gemm128_wmma(const float* __restrict__ X,
                     const float* __restrict__ W, int wrowoff,
                     const float* __restrict__ bias,
                     float* __restrict__ Y, int N, int accum)
{
    __shared__ float As[16 * LDSA];
    const int tid     = threadIdx.x;
    const int wave    = tid >> 5;
    const int lane    = tid & 31;
    const int half    = lane >> 4;   // 0: K lo pair, 1: K hi pair
    const int l       = lane & 15;
    const int rowbase = blockIdx.x * 16;

    // Cooperative, coalesced A-tile load (16 x 128)
    for (int idx = tid; idx < 16 * 128; idx += 256) {
        int r = idx >> 7, c = idx & 127;
        int gr = rowbase + r;
        As[r * LDSA + c] = (gr < N) ? X[(size_t)gr * 128 + c] : 0.0f;
    }
    __syncthreads();

    const int colbase = wave * 16;

    // Single per-thread B base pointer; all K-steps are immediate offsets.
    const float* __restrict__ wbase =
        W + ((size_t)wrowoff + 2 * half) * 128 + colbase + l;
    const float* __restrict__ abase = &As[l * LDSA + 2 * half];

    v8f acc = {};

#pragma unroll
    for (int k0 = 0; k0 < 128; k0 += 4) {
        // A fragment: lane holds row M=l, K = k0+2*half, +1 (8B aligned LDS)
        v2f a = *(const v2f*)(abase + k0);
        // B fragment: lane holds col N=l, K = k0+2*half, +1
        v2f b;
        b.x = wbase[k0 * 128];
        b.y = wbase[k0 * 128 + 128];
        acc = __builtin_amdgcn_wmma_f32_16x16x4_f32(
            /*neg_a=*/false, a, /*neg_b=*/false, b,
            /*c_mod=*/(short)0, acc, /*reuse_a=*/false, /*reuse_b=*/false);
    }

    // Epilogue: VGPR j = row (j + 8*half), col = colbase + l
    const int col = colbase + l;
    const float bv = bias ? bias[col] : 0.0f;
    const int row0 = rowbase + half * 8;
#pragma unroll
    for (int j = 0; j < 8; ++j) {
        int row = row0 + j;
        if (row < N) {
            float v = acc[j] + bv;
            if (accum) v += Y[(size_t)row * 128 + col];
            Y[(size_t)row * 128 + col] = v;
        }
    }
}

// ---------------------------------------------------------------------------
// Row-wise L1 normalize in place: x[r,:] /= max(sum|x[r,:]|, 1e-12)
// ---------------------------------------------------------------------------
__global__ __launch_bounds__(128)
void ga_l1norm(float* __restrict__ x)
{
    __shared__ float red[128];
    const int row = blockIdx.x;
    const int t = threadIdx.x;
    float v = x[(size_t)row * 128 + t];
    red[t] = fabsf(v);
    __syncthreads();
    for (int s = 64; s > 0; s >>= 1) {
        if (t < s) red[t] += red[t + s];
        __syncthreads();
    }
    float denom = fmaxf(red[0], 1e-12f);
    x[(size_t)row * 128 + t] = v / denom;
}

__global__ __launch_bounds__(256)
void ga_fill1(float* __restrict__ p, int n)
{
    int i = blockIdx.x * 256 + threadIdx.x;
    if (i < n) p[i] = 1.0f;
}

__global__ __launch_bounds__(256)
void ga_degcount(const int* __restrict__ dst, float* __restrict__ deg, int E)
{
    int i = blockIdx.x * 256 + threadIdx.x;
    if (i < E) unsafeAtomicAdd(&deg[dst[i]], 1.0f);
}

__global__ __launch_bounds__(256)
void ga_rsqrt(float* __restrict__ p, int n)
{
    int i = blockIdx.x * 256 + threadIdx.x;
    if (i < n) p[i] = rsqrtf(p[i]);   // dinv = deg^-1/2 in place
}

// out[i,f] = h[i,f] * dinv[i]^2 + b[f]   (self-loop term + bias)
__global__ __launch_bounds__(256)
void ga_selfloop(const float* __restrict__ h, const float* __restrict__ dinv,
                 const float* __restrict__ b, float* __restrict__ out, int N)
{
    int i = blockIdx.x * 256 + threadIdx.x;
    if (i < N * 128) {
        int r = i >> 7, f = i & 127;
        float di = dinv[r];
        out[i] = h[i] * di * di + b[f];
    }
}

// out[dst,:] += h[src,:] * dinv[src]*dinv[dst]; one wave32 per edge,
// 4 floats per lane (float4 gather -> 4 HW float atomics)
__global__ __launch_bounds__(256)
void ga_scatter(const float* __restrict__ h, const int* __restrict__ src,
                const int* __restrict__ dst, const float* __restrict__ dinv,
                float* __restrict__ out, int E)
{
    const int warp = threadIdx.x >> 5;
    const int lane = threadIdx.x & 31;
    const int e = blockIdx.x * 8 + warp;
    if (e >= E) return;
    const int s = src[e], d = dst[e];
    const float coef = dinv[s] * dinv[d];
    float4 v = ((const float4*)(h + (size_t)s * 128))[lane];
    float* op = out + (size_t)d * 128 + lane * 4;
    unsafeAtomicAdd(op + 0, v.x * coef);
    unsafeAtomicAdd(op + 1, v.y * coef);
    unsafeAtomicAdd(op + 2, v.z * coef);
    unsafeAtomicAdd(op + 3, v.w * coef);
}

// ---------------------------------------------------------------------------
extern "C" void kernel_launch(void* const* d_in, const int* in_sizes, int n_in,
                              void* d_out, int out_size, void* d_ws, size_t ws_size,
                              hipStream_t stream)
{
    const float* rwr1  = (const float*)d_in[0];
    const float* rwr2  = (const float*)d_in[1];
    const float* x1    = (const float*)d_in[2];
    const float* x2    = (const float*)d_in[3];
    const int*   ei1   = (const int*)d_in[4];
    const int*   ei2   = (const int*)d_in[5];
    const float* lin_W = (const float*)d_in[6];
    const float* lin_b = (const float*)d_in[7];
    const float* c1_W  = (const float*)d_in[8];
    const float* c1_b  = (const float*)d_in[9];
    const float* c2_W  = (const float*)d_in[10];
    const float* c2_b  = (const float*)d_in[11];
    const float* cb_W  = (const float*)d_in[12];
    const float* cb_b  = (const float*)d_in[13];

    const int N = in_sizes[0] / 128;
    const int E = in_sizes[4] / 2;

    float* bufP = (float*)d_ws;                 // pos (post L1)      N*128
    float* bufH = bufP + (size_t)N * 128;       // h = x@W            N*128
    float* bufC = bufH + (size_t)N * 128;       // conv out / g_emb   N*128
    float* bufD = bufC + (size_t)N * 128;       // deg -> dinv        N
    float* out  = (float*)d_out;

    const dim3 gGemm((N + 15) / 16), bGemm(256);
    const dim3 gRow(N), bRow(128);
    const int gNE = (N * 128 + 255) / 256;
    const int gN  = (N + 255) / 256;
    const int gE1 = (E + 255) / 256;
    const int gE8 = (E + 7) / 8;

    for (int g = 0; g < 2; ++g) {
        const float* rwr = g ? rwr2 : rwr1;
        const float* x   = g ? x2 : x1;
        const int* src = g ? ei2 : ei1;
        const int* dst = src + E;
        float* og = out + (size_t)g * N * 128;

        // pos = L1( rwr @ lin_W + lin_b )
        ga_gemm128_wmma<<<gGemm, bGemm, 0, stream>>>(rwr, lin_W, 0, lin_b, bufP, N, 0);
        ga_l1norm<<<gRow, bRow, 0, stream>>>(bufP);

        // dinv = rsqrt(1 + indegree)
        ga_fill1<<<gN, 256, 0, stream>>>(bufD, N);
        ga_degcount<<<gE1, 256, 0, stream>>>(dst, bufD, E);
        ga_rsqrt<<<gN, 256, 0, stream>>>(bufD, N);

        // conv1: h = x@W1 ; c = h/deg + b1 ; c[dst] += h[src]*dinv_s*dinv_d
        ga_gemm128_wmma<<<gGemm, bGemm, 0, stream>>>(x, c1_W, 0, nullptr, bufH, N, 0);
        ga_selfloop<<<gNE, 256, 0, stream>>>(bufH, bufD, c1_b, bufC, N);
        ga_scatter<<<gE8, 256, 0, stream>>>(bufH, src, dst, bufD, bufC, E);

        // conv2 (input bufC, same structure)
        ga_gemm128_wmma<<<gGemm, bGemm, 0, stream>>>(bufC, c2_W, 0, nullptr, bufH, N, 0);
        ga_selfloop<<<gNE, 256, 0, stream>>>(bufH, bufD, c2_b, bufC, N);
        ga_scatter<<<gE8, 256, 0, stream>>>(bufH, src, dst, bufD, bufC, E);
        ga_l1norm<<<gRow, bRow, 0, stream>>>(bufC);

        // emd = L1( [pos, g] @ comb_W + comb_b )  (two K=128 passes, no concat)
        ga_gemm128_wmma<<<gGemm, bGemm, 0, stream>>>(bufP, cb_W, 0, cb_b, og, N, 0);
        ga_gemm128_wmma<<<gGemm, bGemm, 0, stream>>>(bufC, cb_W, 128, nullptr, og, N, 1);
        ga_l1norm<<<gRow, bRow, 0, stream>>>(og);
    }
}